// LiquidEchoHeadV2_75892072121073
// MI455X (gfx1250) — compile-verified
//
#include <hip/hip_runtime.h>
#include <math.h>

// ---------------- Problem constants (match reference) ----------------
#define TPB           256
#define DM            2048
#define ELTS          (DM / TPB)          // 8 columns per thread
#define LUT_N         4096
#define ROWS_PER_BLOCK 8
#define NWAVES        (TPB / 32)

#define PHI_F         1.61803398874989484820f
#define INV_TWO_PI_F  0.15915494309189533577f   // 1/(2*pi)
#define INV_SCALE_F   0.02209708691207961101f   // 1/sqrt(2048)

// LUT sin/cos with linear interpolation; lut holds interleaved (sin[i], cos[i])
// pairs in LDS so each index fetch is one aligned ds_load_b64.
__device__ __forceinline__ void lut_sincos(const float2* __restrict__ lut,
                                           float theta, float& s, float& c) {
    float u = theta * INV_TWO_PI_F;     // theta / (2*pi)
    u -= floorf(u);                     // frac -> [0,1)  (== mod(theta,2pi)/2pi)
    float pos = u * (float)LUT_N;       // [0, LUT_N)
    float i0f = floorf(pos);
    float frac = pos - i0f;
    int i0 = ((int)i0f) & (LUT_N - 1);
    int i1 = (i0 + 1) & (LUT_N - 1);
    float2 a = lut[i0];                 // ds_load_b64 (8B aligned)
    float2 b = lut[i1];                 // ds_load_b64
    float om = 1.0f - frac;
    s = a.x * om + b.x * frac;
    c = a.y * om + b.y * frac;
}

__global__ __launch_bounds__(TPB)
void liquid_echo_kernel(const float* __restrict__ x_real,
                        const float* __restrict__ x_imag,
                        const float* __restrict__ t_vec,
                        const float* __restrict__ mem_real,
                        const float* __restrict__ mem_imag,
                        const float* __restrict__ w_trigger,
                        const float* __restrict__ b_trigger,
                        const float* __restrict__ w_state,
                        const float* __restrict__ b_state,
                        const float* __restrict__ k_ptr,
                        const float* __restrict__ sin_table,
                        const float* __restrict__ cos_table,
                        float* __restrict__ out_real,
                        float* __restrict__ out_imag,
                        int n_rows, int rows_per_block) {
    __shared__ float2 lut[LUT_N];       // interleaved (sin, cos): 32 KiB
    __shared__ float  red[NWAVES];
    __shared__ float  s_alpha;

    const int tid = threadIdx.x;

    // ------------------------------------------------------------------
    // Stage interleaved LUT into LDS using CDNA5 async global->LDS loads.
    // Per-lane LDS destination addresses let us interleave sin/cos for free:
    //   sin_table[i] goes to byte 8*i of lut in LDS, cos_table[i] to byte 8*i + 4
    // Tracked by ASYNCcnt; wave waits, then the workgroup barrier makes the
    // LDS contents visible to all 8 waves.
    // ------------------------------------------------------------------
    {
        unsigned lbase = (unsigned)(size_t)(&lut[0]);   // low 32 bits of the
        // flat LDS-aperture address are the LDS byte offset (ISA 10.2).
        unsigned long long sbase = (unsigned long long)(const void*)sin_table;
        unsigned long long cbase = (unsigned long long)(const void*)cos_table;
        #pragma unroll
        for (int i = tid; i < LUT_N; i += TPB) {
            unsigned ls = lbase + 8u * (unsigned)i;
            unsigned go = 4u * (unsigned)i;
            asm volatile("global_load_async_to_lds_b32 %0, %1, %2"
                         :: "v"(ls), "v"(go), "s"(sbase) : "memory");
            asm volatile("global_load_async_to_lds_b32 %0, %1, %2"
                         :: "v"(ls + 4u), "v"(go), "s"(cbase) : "memory");
        }
        asm volatile("s_wait_asynccnt 0" ::: "memory");
    }
    __syncthreads();

    // ---- Hoist per-column parameters (amortized over rows_per_block rows) ----
    float inv_wlt[ELTS], bt[ELTS], inv_wls[ELTS], bs[ELTS];
    #pragma unroll
    for (int e = 0; e < ELTS; ++e) {
        int j = tid + e * TPB;
        inv_wlt[e] = 1.0f / (1.0f + fabsf(w_trigger[j]));
        bt[e]      = b_trigger[j];
        inv_wls[e] = 1.0f / (1.0f + fabsf(w_state[j]));
        bs[e]      = b_state[j];
    }
    const float k_eff = fabsf(k_ptr[0]) + 0.1f;

    for (int rr = 0; rr < rows_per_block; ++rr) {
        const int row = blockIdx.x * rows_per_block + rr;
        if (row >= n_rows) break;                 // block-uniform
        const size_t base = (size_t)row * DM;
        const float tphi = t_vec[row] * PHI_F;

        // ---- Pass 1: trigger phases + per-row interference partial ----
        float xr[ELTS], xi[ELTS];
        #pragma unroll
        for (int e = 0; e < ELTS; ++e) {
            int j = tid + e * TPB;
            xr[e] = __builtin_nontemporal_load(&x_real[base + j]);
            xi[e] = __builtin_nontemporal_load(&x_imag[base + j]);
        }

        float partial = 0.0f;
        #pragma unroll
        for (int e = 0; e < ELTS; ++e) {
            float thr = xr[e] * inv_wlt[e] + bt[e] + tphi;
            float thi = xi[e] * inv_wlt[e] + bt[e] + tphi;
            float sr, cr, si, ci;
            lut_sincos(lut, thr, sr, cr);
            lut_sincos(lut, thi, si, ci);
            float trig_r = cr * ci - sr * si;
            float trig_i = cr * si + sr * ci;
            partial += trig_r * xr[e] + trig_i * xi[e];
        }

        // ---- wave32 shuffle reduction, then 8-wave LDS combine ----
        #pragma unroll
        for (int m = 16; m >= 1; m >>= 1)
            partial += __shfl_xor(partial, m, 32);
        if ((tid & 31) == 0) red[tid >> 5] = partial;
        __syncthreads();
        if (tid == 0) {
            float s = 0.0f;
            #pragma unroll
            for (int w = 0; w < NWAVES; ++w) s += red[w];
            float ic   = fminf(fmaxf(s * INV_SCALE_F, -1.0f), 1.0f);
            float xinv = (1.0f - ic) * 0.5f;
            s_alpha = expf(-k_eff * xinv);
        }
        __syncthreads();
        const float alpha = s_alpha;
        const float one_m = 1.0f - alpha;

        // ---- Pass 2: blend with memory, re-evolve, write outputs ----
        #pragma unroll
        for (int e = 0; e < ELTS; ++e) {
            int j = tid + e * TPB;
            float mr = __builtin_nontemporal_load(&mem_real[base + j]);
            float mi = __builtin_nontemporal_load(&mem_imag[base + j]);
            float br = alpha * xr[e] + one_m * mr;
            float bi = alpha * xi[e] + one_m * mi;
            float thr = br * inv_wls[e] + bs[e] + tphi;
            float thi = bi * inv_wls[e] + bs[e] + tphi;
            float sr, cr, si, ci;
            lut_sincos(lut, thr, sr, cr);
            lut_sincos(lut, thi, si, ci);
            __builtin_nontemporal_store(cr * ci - sr * si, &out_real[base + j]);
            __builtin_nontemporal_store(cr * si + sr * ci, &out_imag[base + j]);
        }
        // No extra barrier needed: red/s_alpha for the next row are only
        // written after the next row's __syncthreads() pairing.
    }
}

extern "C" void kernel_launch(void* const* d_in, const int* in_sizes, int n_in,
                              void* d_out, int out_size, void* d_ws, size_t ws_size,
                              hipStream_t stream) {
    (void)n_in; (void)out_size; (void)d_ws; (void)ws_size;
    const float* x_real = (const float*)d_in[0];
    const float* x_imag = (const float*)d_in[1];
    const float* t_vec  = (const float*)d_in[2];
    const float* m_real = (const float*)d_in[3];
    const float* m_imag = (const float*)d_in[4];
    const float* w_tr   = (const float*)d_in[5];
    const float* b_tr   = (const float*)d_in[6];
    const float* w_st   = (const float*)d_in[7];
    const float* b_st   = (const float*)d_in[8];
    const float* k_ptr  = (const float*)d_in[9];
    const float* sin_t  = (const float*)d_in[10];
    const float* cos_t  = (const float*)d_in[11];

    const int batch = in_sizes[2];                 // t has BATCH elements
    float* out_real = (float*)d_out;
    float* out_imag = out_real + (size_t)batch * DM;

    const int rpb  = ROWS_PER_BLOCK;
    const int grid = (batch + rpb - 1) / rpb;      // 8192/8 = 1024 blocks
    liquid_echo_kernel<<<grid, TPB, 0, stream>>>(
        x_real, x_imag, t_vec, m_real, m_imag,
        w_tr, b_tr, w_st, b_st, k_ptr, sin_t, cos_t,
        out_real, out_imag, batch, rpb);
}